// GATEncoder_48412871360749
// MI455X (gfx1250) — compile-verified
//
#include <hip/hip_runtime.h>
#include <hip/hip_bf16.h>
#include <cstdint>

typedef float v2f __attribute__((ext_vector_type(2)));
typedef float v8f __attribute__((ext_vector_type(8)));

// ---------- order-preserving float <-> uint for atomicMax-based segment_max ----------
__device__ __forceinline__ unsigned encOrd(float f) {
    unsigned u = __float_as_uint(f);
    return (u & 0x80000000u) ? ~u : (u ^ 0x80000000u);
}
__device__ __forceinline__ float decOrd(unsigned u) {
    unsigned v = (u & 0x80000000u) ? (u ^ 0x80000000u) : ~u;
    return __uint_as_float(v);
}

// =====================================================================
// D[M,Nc] = A[M,K] x B[K,Nc] (+ bias), pure fp32 via V_WMMA_F32_16X16X4_F32.
// One wave per 16x16 output tile; block = 4 waves covering 64 columns.
// Requires M%16==0, K%4==0, Nc%64==0 (true here: 50000/16=3125, 64/16=4).
// All bounds branches are wave-uniform -> EXEC all-1s at WMMA (ISA 7.12).
// Fragment layouts per ISA 7.12.2:
//   A 16x4 : lanes 0-15 M=lane {K=k,k+1}, lanes 16-31 M=lane-16 {K=k+2,k+3}
//   B 4x16 : N=lane&15; VGPR j holds K=k+j (low half) / K=k+j+2 (high half)
//   C 16x16: VGPR v: M = v + 8*(lane>=16), N = lane&15
// =====================================================================
__global__ void gemm_f32_wmma(const float* __restrict__ A,
                              const float* __restrict__ B,
                              float* __restrict__ D,
                              const float* __restrict__ bias,
                              int M, int K, int Nc)
{
    const int lane    = threadIdx.x & 31;
    const int wave    = threadIdx.x >> 5;
    const int rowBase = blockIdx.x << 4;
    const int colBase = (blockIdx.y << 6) + (wave << 4);
    if (rowBase >= M || colBase >= Nc) return;   // wave-uniform

    const int m    = lane & 15;
    const int half = lane >> 4;                  // 0 -> K{0,1}, 1 -> K{2,3}
    const float* arow = A + (size_t)(rowBase + m) * K + (half << 1);
    const float* bcol = B + colBase + m;

    v8f acc = {};
#pragma unroll 8
    for (int k = 0; k < K; k += 4) {
        v2f a, b;
        a.x = arow[k + 0];                       // K = k + 2*half
        a.y = arow[k + 1];                       // K = k + 2*half + 1
        const int kb = k + (half << 1);
        b.x = bcol[(size_t)(kb + 0) * Nc];       // VGPR0: K = k + 2*half
        b.y = bcol[(size_t)(kb + 1) * Nc];       // VGPR1: K = k + 2*half + 1
        acc = __builtin_amdgcn_wmma_f32_16x16x4_f32(
                  false, a, false, b, (short)0, acc, false, false);
    }

    const float bb = bias ? bias[colBase + m] : 0.0f;
#pragma unroll
    for (int v = 0; v < 8; ++v) {
        const int r = v + (half << 3);
        D[(size_t)(rowBase + r) * Nc + colBase + m] = acc[v] + bb;
    }
}

// ---------- per-(node,head) attention logits: al = <h[n,h,:], att> ----------
__global__ void att_logits(const float* __restrict__ h,
                           const float* __restrict__ att_s,
                           const float* __restrict__ att_d,
                           float* __restrict__ al_s,
                           float* __restrict__ al_d,
                           int N, int H, int C)
{
    const int idx = blockIdx.x * blockDim.x + threadIdx.x;
    if (idx >= N * H) return;
    const int node = idx / H, hd = idx - node * H;
    const float4* hp = (const float4*)(h + (size_t)node * H * C + (size_t)hd * C);
    const float4* as = (const float4*)(att_s + (size_t)hd * C);
    const float4* ad = (const float4*)(att_d + (size_t)hd * C);
    float s = 0.f, d = 0.f;
#pragma unroll 4
    for (int c = 0; c < C / 4; ++c) {
        const float4 v = hp[c], a = as[c], b = ad[c];
        s += v.x * a.x + v.y * a.y + v.z * a.z + v.w * a.w;
        d += v.x * b.x + v.y * b.y + v.z * b.z + v.w * b.w;
    }
    al_s[idx] = s;
    al_d[idx] = d;
}

// ---------- edge pass 1: segment max of leaky_relu(al_s[src]+al_d[dst], 0.2) ----------
// Edge list is [real edges | N self-loops].
__global__ void edge_max(const int* __restrict__ src, const int* __restrict__ dst,
                         const float* __restrict__ al_s, const float* __restrict__ al_d,
                         unsigned* __restrict__ emax, int E, int N, int H)
{
    const long long idx = (long long)blockIdx.x * blockDim.x + threadIdx.x;
    const long long total = (long long)(E + N) * H;
    if (idx >= total) return;
    const int e  = (int)(idx / H);
    const int hd = (int)(idx - (long long)e * H);
    int s, d;
    if (e < E) { s = src[e]; d = dst[e]; } else { s = d = e - E; }
    float v = al_s[s * H + hd] + al_d[d * H + hd];
    v = v > 0.f ? v : 0.2f * v;
    atomicMax(&emax[d * H + hd], encOrd(v));
}

// ---------- edge pass 2: w=exp(e-max); acc[dst] += w*h[src]; denom[dst] += w ----------
// One wave per edge; 32 lanes split the H*C channels (vpt = H*C/32 each,
// each lane's chunk stays inside one head since C=64 and vpt in {2,8}).
__global__ void edge_aggregate(const int* __restrict__ src, const int* __restrict__ dst,
                               const float* __restrict__ al_s, const float* __restrict__ al_d,
                               const unsigned* __restrict__ emax,
                               const float* __restrict__ h,
                               float* __restrict__ acc, float* __restrict__ denom,
                               int E, int N, int H, int C)
{
    const int lane = threadIdx.x & 31;
    const long long e = (long long)blockIdx.x * (blockDim.x >> 5) + (threadIdx.x >> 5);
    if (e >= (long long)(E + N)) return;          // wave-uniform
    int s, d;
    if (e < E) { s = src[e]; d = dst[e]; } else { s = d = (int)(e - E); }

    const int elems = H * C;
    const int vpt   = elems >> 5;                 // channels per lane
    const int base  = lane * vpt;
    const int hd    = base / C;

    float eL = al_s[s * H + hd] + al_d[d * H + hd];
    eL = eL > 0.f ? eL : 0.2f * eL;
    const float w = expf(eL - decOrd(emax[d * H + hd]));

    if ((base & (C - 1)) == 0)                    // one lane per head
        atomicAdd(&denom[d * H + hd], w);

    const float* hsrc = h   + (size_t)s * elems + base;
    float*       adst = acc + (size_t)d * elems + base;
#pragma unroll 8
    for (int j = 0; j < vpt; ++j)
        atomicAdd(&adst[j], w * hsrc[j]);
}

// ---------- node finalize: out = leaky( acc/(denom+1e-16) + bias, slope ) ----------
__global__ void node_finalize(const float* __restrict__ acc,
                              const float* __restrict__ denom,
                              const float* __restrict__ bias,
                              float* __restrict__ out,
                              int N, int H, int C, float slope)
{
    const long long idx = (long long)blockIdx.x * blockDim.x + threadIdx.x;
    const long long total = (long long)N * H * C;
    if (idx >= total) return;
    const int rem  = (int)(idx % (H * C));
    const int node = (int)(idx / (H * C));
    const int hd   = rem / C;
    float v = acc[idx] / (denom[node * H + hd] + 1e-16f) + bias[rem];
    out[idx] = v > 0.f ? v : slope * v;
}

// ---------- global mean pool: atomic accumulate per graph ----------
__global__ void pool_accum(const float* __restrict__ h, const int* __restrict__ batch,
                           float* __restrict__ sums, float* __restrict__ cnts,
                           int N, int C)
{
    const long long idx = (long long)blockIdx.x * blockDim.x + threadIdx.x;
    if (idx >= (long long)N * C) return;
    const int node = (int)(idx / C), c = (int)(idx - (long long)node * C);
    const int g = batch[node];
    atomicAdd(&sums[(size_t)g * C + c], h[idx]);
    if (c == 0) atomicAdd(&cnts[g], 1.0f);
}

__global__ void pool_div(const float* __restrict__ sums, const float* __restrict__ cnts,
                         float* __restrict__ pooled, int G, int C)
{
    const int idx = blockIdx.x * blockDim.x + threadIdx.x;
    if (idx >= G * C) return;
    pooled[idx] = sums[idx] / fmaxf(cnts[idx / C], 1.0f);
}

// ---------- grid-stride zero fill (ws is poisoned; accumulators need 0 / enc(-inf)=0) ----------
__global__ void zero_u32(unsigned* __restrict__ p, size_t n)
{
    size_t i = (size_t)blockIdx.x * blockDim.x + threadIdx.x;
    const size_t stride = (size_t)gridDim.x * blockDim.x;
    for (; i < n; i += stride) p[i] = 0u;
}

static inline unsigned ceil_div(long long a, long long b) { return (unsigned)((a + b - 1) / b); }

extern "C" void kernel_launch(void* const* d_in, const int* in_sizes, int n_in,
                              void* d_out, int out_size, void* d_ws, size_t ws_size,
                              hipStream_t stream)
{
    const float* x   = (const float*)d_in[0];
    const int*   ei  = (const int*)  d_in[1];
    const int*   bat = (const int*)  d_in[2];
    const float* W1  = (const float*)d_in[3];
    const float* as1 = (const float*)d_in[4];
    const float* ad1 = (const float*)d_in[5];
    const float* b1  = (const float*)d_in[6];
    const float* W2  = (const float*)d_in[7];
    const float* as2 = (const float*)d_in[8];
    const float* ad2 = (const float*)d_in[9];
    const float* b2  = (const float*)d_in[10];
    const float* lw  = (const float*)d_in[11];
    const float* lb  = (const float*)d_in[12];

    const int N   = in_sizes[2];            // 50000
    const int E   = in_sizes[1] / 2;        // 800000
    const int IN  = in_sizes[0] / N;        // 128
    const int C   = in_sizes[8];            // HID = 64 (att_src2 is [1,HID])
    const int H1  = in_sizes[4] / C;        // 4
    const int OUT = in_sizes[12];           // 64
    const int G   = out_size / OUT;         // 64

    const int* src = ei;
    const int* dst = ei + E;

    // ---- workspace layout: zero-initialized region first, then scratch ----
    float* w = (float*)d_ws;
    size_t p = 0;
    auto alloc = [&](size_t n) { float* r = w + p; p += n; return r; };

    float*    acc1  = alloc((size_t)N * H1 * C);    // layer1 scatter accumulator
    float*    den1  = alloc((size_t)N * H1);        // layer1 softmax denominators
    unsigned* emax1 = (unsigned*)alloc((size_t)N * H1); // enc(-inf) == 0
    float*    acc2  = alloc((size_t)N * C);
    float*    den2  = alloc((size_t)N);
    unsigned* emax2 = (unsigned*)alloc((size_t)N);
    float*    sums  = alloc((size_t)G * C);
    float*    cnts  = alloc((size_t)G);
    const size_t zwords = p;                        // everything above zeroed each call

    float* h1   = alloc((size_t)N * H1 * C);        // x @ W1
    float* als1 = alloc((size_t)N * H1);
    float* ald1 = alloc((size_t)N * H1);
    float* h2   = alloc((size_t)N * C);             // act1 @ W2
    float* als2 = alloc((size_t)N);
    float* ald2 = alloc((size_t)N);
    float* pooled = alloc((size_t)G * C);
    float* act1 = h1;                               // finalize overwrites h1 (dead after aggregate)
    float* act2 = h2;                               // likewise
    (void)ws_size; (void)n_in;

    zero_u32<<<2048, 256, 0, stream>>>((unsigned*)d_ws, zwords);

    // ================= Layer 1 =================
    {
        dim3 grid(ceil_div(N, 16), (unsigned)((H1 * C) / 64));
        gemm_f32_wmma<<<grid, 128, 0, stream>>>(x, W1, h1, nullptr, N, IN, H1 * C);
    }
    att_logits<<<ceil_div((long long)N * H1, 256), 256, 0, stream>>>(
        h1, as1, ad1, als1, ald1, N, H1, C);
    edge_max<<<ceil_div((long long)(E + N) * H1, 256), 256, 0, stream>>>(
        src, dst, als1, ald1, emax1, E, N, H1);
    edge_aggregate<<<ceil_div(E + N, 8), 256, 0, stream>>>(
        src, dst, als1, ald1, emax1, h1, acc1, den1, E, N, H1, C);
    node_finalize<<<ceil_div((long long)N * H1 * C, 256), 256, 0, stream>>>(
        acc1, den1, b1, act1, N, H1, C, 0.01f);

    // ================= Layer 2 (H=1) =================
    {
        dim3 grid(ceil_div(N, 16), (unsigned)(C / 64));
        gemm_f32_wmma<<<grid, 128, 0, stream>>>(act1, W2, h2, nullptr, N, H1 * C, C);
    }
    att_logits<<<ceil_div(N, 256), 256, 0, stream>>>(h2, as2, ad2, als2, ald2, N, 1, C);
    edge_max<<<ceil_div(E + N, 256), 256, 0, stream>>>(src, dst, als2, ald2, emax2, E, N, 1);
    edge_aggregate<<<ceil_div(E + N, 8), 256, 0, stream>>>(
        src, dst, als2, ald2, emax2, h2, acc2, den2, E, N, 1, C);
    node_finalize<<<ceil_div((long long)N * C, 256), 256, 0, stream>>>(
        acc2, den2, b2, act2, N, 1, C, 0.01f);

    // ================= Mean pool + final linear =================
    pool_accum<<<ceil_div((long long)N * C, 256), 256, 0, stream>>>(act2, bat, sums, cnts, N, C);
    pool_div<<<ceil_div(G * C, 256), 256, 0, stream>>>(sums, cnts, pooled, G, C);
    {
        dim3 grid(ceil_div(G, 16), (unsigned)(OUT / 64));
        gemm_f32_wmma<<<grid, 128, 0, stream>>>(pooled, lw, (float*)d_out, lb, G, C, OUT);
    }
}